// PagedAttention_72877005079071
// MI455X (gfx1250) — compile-verified
//
#include <hip/hip_runtime.h>

typedef __attribute__((ext_vector_type(2))) float v2f;
typedef __attribute__((ext_vector_type(8))) float v8f;
typedef __attribute__((ext_vector_type(4))) int   v4i;

#define GLOBAL_AS __attribute__((address_space(1)))
#define LDS_AS    __attribute__((address_space(3)))

typedef GLOBAL_AS v4i* gv4i_p; // global-AS pointer to 16B vector
typedef LDS_AS    v4i* lv4i_p; // LDS-AS pointer to 16B vector

#define NS   32
#define NH   32
#define KVH  8
#define HS   128
#define QPK  4
#define BS   16
#define MB   128
#define NW   4
#define SCALE 0.08838834764831845f /* 1/sqrt(128) */
#define BLKF (HS * BS)             /* 2048 floats per (block, kv-head) slab */

// Async-DMA one 8KB (2048-float) contiguous slab from global to LDS.
// 16 x global_load_async_to_lds_b128 per wave (each covers 32 lanes x 16B).
__device__ __forceinline__ void async_copy_8kb(const float* __restrict__ g,
                                               float* l, int li) {
#pragma unroll
  for (int j = 0; j < 16; ++j) {
    const int e = (j * 32 + li) * 4;
    __builtin_amdgcn_global_load_async_to_lds_b128((gv4i_p)(g + e), (lv4i_p)(l + e),
                                                   0, 0);
  }
}

__launch_bounds__(128)
__global__ void paged_attn_f32wmma_kernel(const float* __restrict__ query,
                                          const float* __restrict__ knew,
                                          const float* __restrict__ vnew,
                                          const float* __restrict__ kcache,
                                          const float* __restrict__ vcache,
                                          const int*   __restrict__ block_tables,
                                          const int*   __restrict__ context_lens,
                                          float*       __restrict__ out) {
  // 128KB: per wave w, buffer b: K at (w*2+b)*4096, V at +2048.
  __shared__ float smem[NW * 2 * 2 * BLKF];
  // Combine-stage overlay (used only after all compute + a barrier):
  float* accS = smem;          // [NW][32][64] = 8192 floats
  float* mS   = smem + 8192;   // [NW][32]
  float* lS   = smem + 8320;   // [NW][32]
  float* outS = smem + 8448;   // [512]

  const int wg  = blockIdx.x;
  const int seq = wg >> 3;
  const int kvh = wg & 7;
  const int tid = threadIdx.x;
  const int w   = tid >> 5;   // wave id 0..3
  const int li  = tid & 31;   // lane id
  const int q4  = li & 15;    // column (query head) / token-in-block index
  const bool lower = li < 16;
  const int koff = lower ? 0 : 2; // K-dim sub-chunk held by this lane half
  const int hi8  = lower ? 0 : 8; // score-row offset of this lane half

  const int ctx     = context_lens[seq];
  const int last    = ctx - 1;               // new-token position (== slot_mapping)
  const int nblocks = (ctx + BS - 1) / BS;

  // ---- preload Q (B-operand of QK^T), pre-scaled; zero for padded columns ----
  const float qsel = (q4 < QPK) ? SCALE : 0.0f;
  const int   qr   = (q4 < QPK) ? q4 : (QPK - 1);
  const float* qrow = query + (size_t)seq * (NH * HS) + (size_t)(kvh * QPK + qr) * HS;
  v2f qb[32];
#pragma unroll
  for (int c = 0; c < 32; ++c) {
    v2f t = *(const v2f*)(qrow + 4 * c + koff);
    qb[c].x = t.x * qsel;
    qb[c].y = t.y * qsel;
  }

  const float* knew_row = knew + (size_t)(seq * KVH + kvh) * HS;
  const float* vnew_row = vnew + (size_t)(seq * KVH + kvh) * HS;
  float* stage = smem + w * (2 * 2 * 2048); // this wave's 2 x 16KB buffers

  float m = -3.0e38f;
  float l = 0.0f;
  v8f acc[8]; // acc[dt] holds d = dt*16 + {r | r+8}, column q4
#pragma unroll
  for (int dt = 0; dt < 8; ++dt) acc[dt] = (v8f){0, 0, 0, 0, 0, 0, 0, 0};

  // ---- prologue: DMA first block into buffer 0 ----
  if (w < nblocks) {
    const int blk0 = __builtin_amdgcn_readfirstlane(block_tables[seq * MB + w]);
    const float* kb0 = kcache + (size_t)(blk0 * KVH + kvh) * BLKF;
    const float* vb0 = vcache + (size_t)(blk0 * KVH + kvh) * BLKF;
    async_copy_8kb(kb0, stage, li);
    async_copy_8kb(vb0, stage + 2048, li);
  }

  int buf = 0;
  for (int bi = w; bi < nblocks; bi += NW, buf ^= 1) {
    const int nxt = bi + NW;
    if (nxt < nblocks) {
      // prefetch next block into the other buffer, then wait for current (<=32 in flight)
      const int nblk = __builtin_amdgcn_readfirstlane(block_tables[seq * MB + nxt]);
      const float* kbn = kcache + (size_t)(nblk * KVH + kvh) * BLKF;
      const float* vbn = vcache + (size_t)(nblk * KVH + kvh) * BLKF;
      float* nstage = stage + (buf ^ 1) * 4096;
      async_copy_8kb(kbn, nstage, li);
      async_copy_8kb(vbn, nstage + 2048, li);
      __builtin_amdgcn_s_wait_asynccnt(32);
    } else {
      __builtin_amdgcn_s_wait_asynccnt(0);
    }

    float* kl = stage + buf * 4096;
    float* vl = kl + 2048;
    const int tokBase = bi * BS;

    // ---- reshape_and_cache: patch the new token's K/V row into the LDS tile ----
    if (bi == (last >> 4)) {           // wave-uniform
      const int s = last & 15;
#pragma unroll
      for (int j = 0; j < 4; ++j) {
        const int d = li * 4 + j;
        kl[(d >> 3) * 128 + s * 8 + (d & 7)] = knew_row[d];
        vl[d * BS + s] = vnew_row[d];
      }
    }

    // ---------- scores S[t][q]: 32 x v_wmma_f32_16x16x4_f32, operands from LDS ----------
    v8f S = (v8f){0, 0, 0, 0, 0, 0, 0, 0};
#pragma unroll
    for (int c = 0; c < 32; ++c) {
      const int d = 4 * c + koff;
      const v2f ka = *(const v2f*)(kl + ((d >> 3) * 128 + q4 * 8 + (d & 7)));
      S = __builtin_amdgcn_wmma_f32_16x16x4_f32(false, ka, false, qb[c], (short)0, S,
                                                false, false);
    }

    // ---------- online softmax (column q4, halves synced via shfl_xor) ----------
    float p[8];
    float tm = -3.0e38f;
#pragma unroll
    for (int r = 0; r < 8; ++r) {
      const float sv = ((tokBase + hi8 + r) < ctx) ? S[r] : -3.0e38f;
      p[r] = sv;
      tm = fmaxf(tm, sv);
    }
    tm = fmaxf(tm, __shfl_xor(tm, 16, 32));
    const float mn = fmaxf(m, tm);
    const float sc = __expf(m - mn);
    m = mn;
    float ps = 0.0f;
#pragma unroll
    for (int r = 0; r < 8; ++r) {
      p[r] = __expf(p[r] - mn); // masked entries -> exp(-huge) = 0
      ps += p[r];
    }
    l = l * sc + ps;
#pragma unroll
    for (int dt = 0; dt < 8; ++dt) acc[dt] *= sc;

    // ---------- out[d][q] += V^T x P : 32 x v_wmma_f32_16x16x4_f32 ----------
#pragma unroll
    for (int cc = 0; cc < 4; ++cc) {
      if (tokBase + cc * 4 < ctx) { // wave-uniform chunk skip
        const int baseR = (cc & 1) * 4;
        const float a0 = p[baseR + 0], a1 = p[baseR + 1];
        const float a2 = p[baseR + 2], a3 = p[baseR + 3];
        const float u0 = __shfl(a0, q4 + 16, 32);
        const float u1 = __shfl(a1, q4 + 16, 32);
        const float l2 = __shfl(a2, q4, 32);
        const float l3 = __shfl(a3, q4, 32);
        v2f bp;
        if (cc < 2) { bp.x = lower ? a0 : l2; bp.y = lower ? a1 : l3; }
        else        { bp.x = lower ? u0 : a2; bp.y = lower ? u1 : a3; }
        const int sv = cc * 4 + koff;
#pragma unroll
        for (int dt = 0; dt < 8; ++dt) {
          const int d = dt * 16 + q4;
          const v2f va = *(const v2f*)(vl + d * BS + sv);
          acc[dt] = __builtin_amdgcn_wmma_f32_16x16x4_f32(false, va, false, bp, (short)0,
                                                          acc[dt], false, false);
        }
      }
    }
  }

  __builtin_amdgcn_s_wait_asynccnt(0); // all staging traffic retired
  __syncthreads();                     // staging region now reusable as combine overlay

  // ---------- cross-wave flash-decoding combine ----------
  mS[w * 32 + li] = m;
  lS[w * 32 + li] = l;
  {
    float* as = accS + (w * 32 + li) * 64;
#pragma unroll
    for (int dt = 0; dt < 8; ++dt)
#pragma unroll
      for (int r = 0; r < 8; ++r) as[dt * 8 + r] = acc[dt][r];
  }
  __syncthreads();

  if (tid < 32) {
    float mg = mS[li];
#pragma unroll
    for (int ww = 1; ww < NW; ++ww) mg = fmaxf(mg, mS[ww * 32 + li]);
    float wgt[NW];
    float L = 0.0f;
#pragma unroll
    for (int ww = 0; ww < NW; ++ww) {
      wgt[ww] = __expf(mS[ww * 32 + li] - mg);
      L += wgt[ww] * (lS[ww * 32 + li] + lS[ww * 32 + (li ^ 16)]);
    }
    if (q4 < QPK) {
      const float invL = 1.0f / L;
#pragma unroll
      for (int k = 0; k < 64; ++k) {
        float v = 0.0f;
#pragma unroll
        for (int ww = 0; ww < NW; ++ww) v += wgt[ww] * accS[(ww * 32 + li) * 64 + k];
        const int dt = k >> 3, r = k & 7;
        outS[q4 * HS + dt * 16 + r + hi8] = v * invL;
      }
    }
  }
  __syncthreads();

  // coalesced float4 store: 128 threads x 16B = this WG's 512-float output slab
  const float4 vv = ((const float4*)outS)[tid];
  ((float4*)(out + (size_t)(seq * KVH + kvh) * (QPK * HS)))[tid] = vv;
}

extern "C" void kernel_launch(void* const* d_in, const int* in_sizes, int n_in,
                              void* d_out, int out_size, void* d_ws, size_t ws_size,
                              hipStream_t stream) {
  const float* query = (const float*)d_in[0];
  const float* key   = (const float*)d_in[1];
  const float* value = (const float*)d_in[2];
  const float* kc    = (const float*)d_in[3];
  const float* vc    = (const float*)d_in[4];
  const int*   bt    = (const int*)d_in[5];
  const int*   cl    = (const int*)d_in[6];
  // d_in[7] (slot_mapping) is derivable from block_tables + context_lens; unused.
  (void)in_sizes; (void)n_in; (void)out_size; (void)d_ws; (void)ws_size;

  paged_attn_f32wmma_kernel<<<NS * KVH, NW * 32, 0, stream>>>(
      query, key, value, kc, vc, bt, cl, (float*)d_out);
}